// SeasonalFreqEnhancer_70377334112421
// MI455X (gfx1250) — compile-verified
//
#include <hip/hip_runtime.h>

// ---------------------------------------------------------------------------
// SeasonalFreqEnhancer for MI455X (gfx1250, wave32).
//
// pred = irfft( scale_topk(|rfft(x)|) * rfft(x) ), per row of 4096.
// FFT4096 = three stages of DFT-16 expressed as complex (16x16)x(16x256)
// matmuls on the fp32 WMMA pipe (V_WMMA_F32_16X16X4_F32, K chained 4x),
// with twiddle multiplies between stages. One workgroup per row, in-place
// in LDS (single 4096-pt complex buffer).
//
// Twiddles are linear in the output row, so each tile needs only 2 sincos
// (base + increment) and a complex-product recurrence over the 8 rows,
// keeping the kernel memory-bound (~45us HBM floor) instead of
// transcendental-bound.
// ---------------------------------------------------------------------------

typedef float v2f __attribute__((ext_vector_type(2)));
typedef float v8f __attribute__((ext_vector_type(8)));

#define NFFT     4096
#define NBINS    2049
#define TOPK     204
#define NTHREADS 256
#define PI2      6.28318530717958647692f

__device__ __forceinline__ v8f wmma4(v2f a, v2f b, v8f c) {
  // D(16x16,f32) = A(16x4,f32) * B(4x16,f32) + C ; 8-arg VOP3P form:
  // (neg_a, A, neg_b, B, c_mod, C, reuse_a, reuse_b)
  return __builtin_amdgcn_wmma_f32_16x16x4_f32(false, a, false, b, (short)0, c,
                                               false, false);
}

// N = 16*256 Cooley-Tukey index maps (t = column 0..255 of the 16x256 operand)
__device__ __forceinline__ int idxLoad(int st, int a, int t) {
  if (st == 0) return (a << 8) + t;                               // x[256a + t]
  if (st == 1) return ((t >> 4) << 8) + (a << 4) + (t & 15);      // y[k1*256+16a+b]
  return ((t >> 4) << 8) + ((t & 15) << 4) + a;                   // z[k1*256+16c+b]
}
__device__ __forceinline__ int idxStore(int st, int c, int t) {
  if (st == 0) return (c << 8) + t;                               // y[k1*256+q]
  if (st == 1) return ((t >> 4) << 8) + (c << 4) + (t & 15);      // z[k1*256+16c+b]
  return (t >> 4) + ((t & 15) << 4) + (c << 8);                   // X[k1+16c+256d]
}
// Twiddle angle per unit of output row c (angle is linear in c).
__device__ __forceinline__ float twSlope(int st, int t, float sgn) {
  if (st == 0) return sgn * (PI2 / 4096.0f) * (float)t;
  if (st == 1) return sgn * (PI2 / 256.0f) * (float)(t & 15);
  return 0.0f;
}

// Build the W16 DFT matrix in WMMA A-operand layout (16x4 chunk ck):
// lane(l): row m = l&15 ; lanes 0-15 hold K = 4ck+{0,1}, lanes 16-31 K = 4ck+{2,3}.
__device__ __forceinline__ void makeW(float sgn, v2f* Wr, v2f* Wi, v2f* nWi,
                                      int lane) {
  const int m = lane & 15, hf = lane >> 4;
#pragma unroll
  for (int ck = 0; ck < 4; ++ck) {
    const int k0 = 4 * ck + 2 * hf;
    float s0, c0, s1, c1;
    __sincosf(sgn * (PI2 / 16.0f) * (float)(m * k0), &s0, &c0);
    __sincosf(sgn * (PI2 / 16.0f) * (float)(m * (k0 + 1)), &s1, &c1);
    v2f wr = {c0, c1}, wi = {s0, s1}, nw = {-s0, -s1};
    Wr[ck] = wr; Wi[ck] = wi; nWi[ck] = nw;
  }
}

// One in-place FFT stage: load this wave's two 16-column tiles into registers,
// barrier, then complex matmul (WMMA) + twiddle + scatter store.
// FIRST_REAL: input imaginary part is identically zero (forward stage 0).
// LAST_REAL : only the real output is produced, no twiddle (inverse stage 2).
template <bool FIRST_REAL, bool LAST_REAL>
__device__ __forceinline__ void fft_stage(int st, float sgn, float* re,
                                          float* im, const v2f* Wr,
                                          const v2f* Wi, const v2f* nWi,
                                          int lane, int wv, float scale) {
  const int n = lane & 15, hf = lane >> 4;
  v2f Zr[2][4], Zi[2][4];
#pragma unroll
  for (int tt = 0; tt < 2; ++tt) {
    const int t = ((wv * 2 + tt) << 4) + n;
#pragma unroll
    for (int ck = 0; ck < 4; ++ck) {
      // B-operand 4x16 layout mirrors A: lanes 0-15 K=4ck+{0,1}, 16-31 K=4ck+{2,3}
      const int r0 = 4 * ck + 2 * hf;
      const int i0 = idxLoad(st, r0, t), i1 = idxLoad(st, r0 + 1, t);
      v2f zr = {re[i0], re[i1]};
      Zr[tt][ck] = zr;
      if (!FIRST_REAL) {
        v2f zi = {im[i0], im[i1]};
        Zi[tt][ck] = zi;
      }
    }
  }
  __syncthreads();  // all reads of the in-place buffer done before any write
#pragma unroll
  for (int tt = 0; tt < 2; ++tt) {
    const int t = ((wv * 2 + tt) << 4) + n;
    v8f ar = {0, 0, 0, 0, 0, 0, 0, 0};
    v8f ai = {0, 0, 0, 0, 0, 0, 0, 0};
#pragma unroll
    for (int ck = 0; ck < 4; ++ck) ar = wmma4(Wr[ck], Zr[tt][ck], ar);
    if (!FIRST_REAL) {
#pragma unroll
      for (int ck = 0; ck < 4; ++ck) ar = wmma4(nWi[ck], Zi[tt][ck], ar);
    }
    if (!LAST_REAL) {
#pragma unroll
      for (int ck = 0; ck < 4; ++ck) ai = wmma4(Wi[ck], Zr[tt][ck], ai);
      if (!FIRST_REAL) {
#pragma unroll
        for (int ck = 0; ck < 4; ++ck) ai = wmma4(Wr[ck], Zi[tt][ck], ai);
      }
    }
    // Twiddle e^{i*slope*c}: evaluate at c = 8*hf, then recur over the 8 rows.
    float bc = 1.0f, bs = 0.0f, ic = 1.0f, is = 0.0f;
    if (!LAST_REAL && st != 2) {
      const float slope = twSlope(st, t, sgn);
      __sincosf(slope * (float)(hf << 3), &bs, &bc);
      __sincosf(slope, &is, &ic);
    }
    // C/D layout: VGPR r, lanes 0-15 -> M=r, lanes 16-31 -> M=r+8, N=lane&15
#pragma unroll
    for (int r = 0; r < 8; ++r) {
      const int m = r + (hf << 3);
      const float dr = ar[r];
      if (LAST_REAL) {
        re[idxStore(2, m, t)] = dr * scale;  // real output, in place
      } else {
        const float di = ai[r];
        const int o = idxStore(st, m, t);
        re[o] = dr * bc - di * bs;
        im[o] = dr * bs + di * bc;
        const float nbc = bc * ic - bs * is;  // advance twiddle by one row
        const float nbs = bc * is + bs * ic;
        bc = nbc; bs = nbs;
      }
    }
  }
}

__global__ __launch_bounds__(NTHREADS) void seasonal_freq_enhancer_kernel(
    const float* __restrict__ x, const float* __restrict__ W1,
    const float* __restrict__ b1, const float* __restrict__ W2,
    const float* __restrict__ b2, float* __restrict__ out) {
  __shared__ float Xre[NFFT];
  __shared__ float Xim[NFFT];
  __shared__ float amp[NBINS];
  __shared__ float sc[NBINS];
  __shared__ unsigned hist[NTHREADS];
  __shared__ unsigned sh_pref, sh_rem;
  __shared__ float sW1[16], sb1[16], sW2[16], sb2;

  const int tid = threadIdx.x;
  const int lane = tid & 31;
  const int wv = tid >> 5;
  const long long row = blockIdx.x;
  const float* __restrict__ xin = x + row * (long long)NFFT;
  float* __restrict__ orow = out + row * (long long)NFFT;

  if (tid < 16) { sW1[tid] = W1[tid]; sb1[tid] = b1[tid]; sW2[tid] = W2[tid]; }
  if (tid == 0) sb2 = b2[0];

  // Load the row (coalesced float4), imag implicitly zero via FIRST_REAL stage.
#pragma unroll
  for (int base = 0; base < NFFT; base += NTHREADS * 4) {
    const int i = base + tid * 4;
    const float4 v = *(const float4*)(xin + i);
    Xre[i] = v.x; Xre[i + 1] = v.y; Xre[i + 2] = v.z; Xre[i + 3] = v.w;
  }
  __syncthreads();

  // Forward W16 (sign -1). Inverse W16 is the conjugate: swap Wi <-> nWi.
  v2f Wr[4], Wi[4], nWi[4];
  makeW(-1.0f, Wr, Wi, nWi, lane);

  // ---------------- forward FFT (sign = -1), in place ----------------
  fft_stage<true, false>(0, -1.0f, Xre, Xim, Wr, Wi, nWi, lane, wv, 1.0f);
  __syncthreads();
  fft_stage<false, false>(1, -1.0f, Xre, Xim, Wr, Wi, nWi, lane, wv, 1.0f);
  __syncthreads();
  fft_stage<false, false>(2, -1.0f, Xre, Xim, Wr, Wi, nWi, lane, wv, 1.0f);
  __syncthreads();

  // ---------------- amplitudes of the half spectrum ----------------
  for (int j = tid; j < NBINS; j += NTHREADS) {
    const float re = Xre[j], im = Xim[j];
    amp[j] = sqrtf(re * re + im * im);
  }
  __syncthreads();

  // ---------------- exact top-K threshold via 4x8-bit radix select -------
  // amp >= 0, so the IEEE bit pattern is order-preserving.
  unsigned prefix = 0;
  int remaining = TOPK;
  for (int shift = 24; shift >= 0; shift -= 8) {
    hist[tid] = 0u;
    __syncthreads();
    for (int j = tid; j < NBINS; j += NTHREADS) {
      const unsigned u = __float_as_uint(amp[j]);
      if (shift == 24 || (u >> (shift + 8)) == prefix)
        atomicAdd(&hist[(u >> shift) & 0xffu], 1u);
    }
    __syncthreads();
    if (tid == 0) {
      int cum = 0, b = 255;
      for (; b > 0; --b) {
        cum += (int)hist[b];
        if (cum >= remaining) break;
      }
      if (cum < remaining) { cum += (int)hist[0]; b = 0; }
      sh_rem = (unsigned)(remaining - (cum - (int)hist[b]));
      sh_pref = (prefix << 8) | (unsigned)b;
    }
    __syncthreads();
    prefix = sh_pref;
    remaining = (int)sh_rem;
    __syncthreads();
  }
  const unsigned T = prefix;       // bit pattern of the K-th largest amplitude
  const int need_eq = remaining;   // how many == T to take (smallest index first)

  // ---------------- selection + MLP scale factor per bin ----------------
  for (int j = tid; j < NBINS; j += NTHREADS) {
    const float a = amp[j];
    const unsigned u = __float_as_uint(a);
    bool sel = (u > T);
    if (!sel && u == T) {
      int rank = 0;
      for (int i = 0; i < j; ++i)
        if (__float_as_uint(amp[i]) == T) ++rank;
      sel = (rank < need_eq);
    }
    float s = 0.0f;
    if (sel && a > 0.0f) {
      float accm = sb2;
#pragma unroll
      for (int h = 0; h < 16; ++h)
        accm += fmaxf(fmaf(a, sW1[h], sb1[h]), 0.0f) * sW2[h];
      s = accm / a;  // enhanced amplitude with original phase == s * f[j]
    }
    sc[j] = s;
  }
  __syncthreads();

  // ------- build sparse Hermitian full spectrum in place (pairwise) -------
  if (tid == 0) {
    Xre[0] = sc[0] * Xre[0];           Xim[0] = 0.0f;     // irfft drops DC imag
    Xre[2048] = sc[2048] * Xre[2048];  Xim[2048] = 0.0f;  // and Nyquist imag
  }
  for (int j = tid; j < 2048; j += NTHREADS) {
    if (j == 0) continue;
    const float s = sc[j];
    const float nr = s * Xre[j], ni = s * Xim[j];
    Xre[j] = nr;        Xim[j] = ni;
    Xre[NFFT - j] = nr; Xim[NFFT - j] = -ni;  // conjugate mirror
  }
  __syncthreads();

  // ------------- inverse FFT (sign = +1), real output, 1/N ---------------
  // Inverse W16 = conj(forward W16): pass (Wr, -Wi_fwd, +Wi_fwd) = (Wr, nWi, Wi).
  fft_stage<false, false>(0, 1.0f, Xre, Xim, Wr, nWi, Wi, lane, wv, 1.0f);
  __syncthreads();
  fft_stage<false, false>(1, 1.0f, Xre, Xim, Wr, nWi, Wi, lane, wv, 1.0f);
  __syncthreads();
  fft_stage<false, true>(2, 1.0f, Xre, Xim, Wr, nWi, Wi, lane, wv,
                         1.0f / (float)NFFT);
  __syncthreads();

  // ---------------- coalesced copy-out ----------------
#pragma unroll
  for (int base = 0; base < NFFT; base += NTHREADS * 4) {
    const int i = base + tid * 4;
    float4 v;
    v.x = Xre[i]; v.y = Xre[i + 1]; v.z = Xre[i + 2]; v.w = Xre[i + 3];
    *(float4*)(orow + i) = v;
  }
}

extern "C" void kernel_launch(void* const* d_in, const int* in_sizes, int n_in,
                              void* d_out, int out_size, void* d_ws,
                              size_t ws_size, hipStream_t stream) {
  (void)in_sizes; (void)n_in; (void)out_size; (void)d_ws; (void)ws_size;
  const float* seasonal = (const float*)d_in[0];
  const float* W1 = (const float*)d_in[1];
  const float* b1 = (const float*)d_in[2];
  const float* W2 = (const float*)d_in[3];
  const float* b2 = (const float*)d_in[4];
  float* out = (float*)d_out;

  const int rows = 32 * 1024;  // B*N
  seasonal_freq_enhancer_kernel<<<dim3(rows), dim3(NTHREADS), 0, stream>>>(
      seasonal, W1, b1, W2, b2, out);
}